// GCN_3l_24970939859424
// MI455X (gfx1250) — compile-verified
//
#include <hip/hip_runtime.h>

typedef float v2f __attribute__((ext_vector_type(2)));
typedef float v8f __attribute__((ext_vector_type(8)));

#define NN 50000
#define HID 64

// ---------------- elementwise helpers ----------------

__global__ void fill_f32(float* __restrict__ p, float v, int n) {
    int i = blockIdx.x * blockDim.x + threadIdx.x;
    if (i < n) p[i] = v;
}

__global__ void deg_accum(const int* __restrict__ dst, float* __restrict__ deg, int E) {
    int e = blockIdx.x * blockDim.x + threadIdx.x;
    if (e < E) atomicAdd(&deg[dst[e]], 1.0f);
}

__global__ void rsqrt_inplace(float* __restrict__ p, int n) {
    int i = blockIdx.x * blockDim.x + threadIdx.x;
    if (i < n) p[i] = rsqrtf(p[i]);
}

__global__ void edge_norm(const int* __restrict__ src, const int* __restrict__ dst,
                          const float* __restrict__ dis, float* __restrict__ nrm, int E) {
    int e = blockIdx.x * blockDim.x + threadIdx.x;
    if (e < E) nrm[e] = dis[src[e]] * dis[dst[e]];
}

// Zero-pad Wf2 (K x Nin) into Wp (K x Npad)
__global__ void pad_w(const float* __restrict__ W, float* __restrict__ Wp,
                      int K, int Nin, int Npad) {
    int t = blockIdx.x * blockDim.x + threadIdx.x;
    if (t < K * Npad) {
        int k = t / Npad, n = t % Npad;
        Wp[t] = (n < Nin) ? W[k * Nin + n] : 0.0f;
    }
}

// ---------------- f32 WMMA GEMM: Out[N x Ncols] = X[N x K] @ W[K x ldW] (+bias)(+relu) ----------
// One wave (32 threads) computes one 16x16 output tile using V_WMMA_F32_16X16X4_F32.
// A layout (16x4, MxK): lanes 0-15 -> K=0,1 in vgpr0/1; lanes 16-31 -> K=2,3.
// B layout (4x16, KxN): B^T treated like A: lane holds column n, vgpr j holds K = 2*half + j.
// C/D layout: vgpr j -> M = j (lanes 0-15) / M = j+8 (lanes 16-31), N = lane%16.
__global__ void gemm_wmma(const float* __restrict__ X, const float* __restrict__ W,
                          const float* __restrict__ bias, float* __restrict__ Out,
                          int K, int ldW, int Ncols, int ldOut, int relu) {
    const int lane = threadIdx.x;        // 0..31, exactly one wave per block
    const int half = lane >> 4;          // 0 or 1
    const int l    = lane & 15;
    const int m0   = blockIdx.x * 16;
    const int n0   = blockIdx.y * 16;
    const int n    = n0 + l;

    const float* __restrict__ arow = X + (size_t)(m0 + l) * K;

    v8f acc = {};
    #pragma unroll 4
    for (int k = 0; k < K; k += 4) {
        const int ka = k + half * 2;
        v2f a, b;
        a.x = arow[ka];
        a.y = arow[ka + 1];
        b.x = W[(size_t)ka * ldW + n];        // ldW padded to >= n0+16, always in-bounds
        b.y = W[(size_t)(ka + 1) * ldW + n];
        acc = __builtin_amdgcn_wmma_f32_16x16x4_f32(
                  false, a, false, b, (short)0, acc, false, false);
    }

    if (n < Ncols) {
        const float bv = bias ? bias[n] : 0.0f;
        #pragma unroll
        for (int j = 0; j < 8; ++j) {
            const int m = m0 + half * 8 + j;
            float v = acc[j] + bv;
            if (relu) v = fmaxf(v, 0.0f);
            Out[(size_t)m * ldOut + n] = v;
        }
    }
}

// ---------------- edge scatter: agg[dst] += norm * h[src] ----------------
// 64 consecutive threads = one edge's 64 features -> fully coalesced 256B gather
// + 64 contiguous global_atomic_add_f32 that land in L2 (agg is 12.8MB, L2 is 192MB).
__global__ void edge_scatter(const float* __restrict__ h, const float* __restrict__ nrm,
                             const int* __restrict__ src, const int* __restrict__ dst,
                             float* __restrict__ agg, int E) {
    long long t = (long long)blockIdx.x * blockDim.x + threadIdx.x;
    int e = (int)(t >> 6);
    int f = (int)(t & 63);
    if (e < E) {
        int s = src[e], d = dst[e];
        float v = h[(size_t)s * HID + f] * nrm[e];
        atomicAdd(&agg[(size_t)d * HID + f], v);
    }
}

// ---------------- combine: out = relu(agg + dis^2 * h + b) ----------------
__global__ void combine(const float* __restrict__ agg, const float* __restrict__ h,
                        const float* __restrict__ dis, const float* __restrict__ b,
                        float* __restrict__ out, int N) {
    long long t = (long long)blockIdx.x * blockDim.x + threadIdx.x;
    if (t < (long long)N * HID) {
        int i = (int)(t >> 6);
        int f = (int)(t & 63);
        float di = dis[i];
        out[t] = fmaxf(agg[t] + di * di * h[t] + b[f], 0.0f);
    }
}

// ---------------- host orchestration ----------------

extern "C" void kernel_launch(void* const* d_in, const int* in_sizes, int n_in,
                              void* d_out, int out_size, void* d_ws, size_t ws_size,
                              hipStream_t stream) {
    const float* x   = (const float*)d_in[0];
    const int*   ei  = (const int*)d_in[1];   // edge_index [2, E] (int32 per harness convention)
    const float* W1  = (const float*)d_in[2];
    const float* b1  = (const float*)d_in[3];
    const float* W2  = (const float*)d_in[4];
    const float* b2  = (const float*)d_in[5];
    const float* W3  = (const float*)d_in[6];
    const float* b3  = (const float*)d_in[7];
    const float* Wf1 = (const float*)d_in[8];
    const float* bf1 = (const float*)d_in[9];
    const float* Wf2 = (const float*)d_in[10];
    const float* bf2 = (const float*)d_in[11];

    const int E   = in_sizes[1] / 2;
    const int N   = NN;
    const int Din = in_sizes[0] / N;          // 128
    const int* src = ei;
    const int* dst = ei + E;

    // workspace layout (floats): dis[N] | norm[E] | h[N*64] | agg[N*64] | act[N*64] | wpad[64*16]
    float* ws   = (float*)d_ws;
    float* dis  = ws;                 ws += N;
    float* nrm  = ws;                 ws += E;
    float* hbuf = ws;                 ws += (size_t)N * HID;
    float* agg  = ws;                 ws += (size_t)N * HID;
    float* act  = ws;                 ws += (size_t)N * HID;
    float* wpad = ws;                 ws += HID * 16;

    // --- degree / norm precompute (deg = 1 + segment_sum over dst) ---
    fill_f32<<<(N + 255) / 256, 256, 0, stream>>>(dis, 1.0f, N);
    deg_accum<<<(E + 255) / 256, 256, 0, stream>>>(dst, dis, E);
    rsqrt_inplace<<<(N + 255) / 256, 256, 0, stream>>>(dis, N);
    edge_norm<<<(E + 255) / 256, 256, 0, stream>>>(src, dst, dis, nrm, E);

    const long long NF = (long long)N * HID;
    const int nfBlocks = (int)((NF + 255) / 256);
    const long long EF = (long long)E * HID;
    const int efBlocks = (int)((EF + 255) / 256);
    const dim3 waveBlk(32);
    const dim3 gemmGrid(N / 16, HID / 16);    // 3125 x 4 tiles (50000 = 3125*16 exactly)

    // --- three GCN conv layers ---
    const float* in = x;
    int K = Din;
    const float* Ws[3] = {W1, W2, W3};
    const float* bs[3] = {b1, b2, b3};
    for (int layer = 0; layer < 3; ++layer) {
        gemm_wmma<<<gemmGrid, waveBlk, 0, stream>>>(in, Ws[layer], nullptr, hbuf,
                                                    K, HID, HID, HID, /*relu=*/0);
        fill_f32<<<nfBlocks, 256, 0, stream>>>(agg, 0.0f, (int)NF);
        edge_scatter<<<efBlocks, 256, 0, stream>>>(hbuf, nrm, src, dst, agg, E);
        combine<<<nfBlocks, 256, 0, stream>>>(agg, hbuf, dis, bs[layer], act, N);
        in = act;
        K = HID;
    }

    // --- FFN: relu(act @ Wf1 + bf1) @ Wf2 + bf2 ---
    gemm_wmma<<<gemmGrid, waveBlk, 0, stream>>>(act, Wf1, bf1, hbuf,
                                                HID, HID, HID, HID, /*relu=*/1);
    pad_w<<<(HID * 16 + 255) / 256, 256, 0, stream>>>(Wf2, wpad, HID, 10, 16);
    gemm_wmma<<<dim3(N / 16, 1), waveBlk, 0, stream>>>(hbuf, wpad, bf2, (float*)d_out,
                                                       HID, 16, 10, 10, /*relu=*/0);
}